// SimpleMatcher_82557861364101
// MI455X (gfx1250) — compile-verified
//
#include <hip/hip_runtime.h>
#include <hip/hip_bf16.h>
#include <stdint.h>

#define NB   128   // batches
#define NP   900   // predictions per batch
#define NT   300   // targets per batch
#define TGT_PER_BLK 64
#define THREADS 256
#define PRED_CHUNK (NP / 4)   // 225, each of 4 sub-lanes scans one chunk

#if defined(__has_builtin)
#if __has_builtin(__builtin_amdgcn_global_load_async_to_lds_b128)
#define HAVE_ASYNC_LDS 1
#endif
#endif

// Builtin wants pointer-to-16B-vector operands (per hipcc diagnostic).
typedef int v4i __attribute__((vector_size(16)));
typedef __attribute__((address_space(1))) v4i as1_v4i;
typedef __attribute__((address_space(3))) v4i as3_v4i;

__global__ __launch_bounds__(THREADS)
void giou_argmax_kernel(const float* __restrict__ pred,      // [B,N,4] cxcywh
                        const float* __restrict__ targ,      // [B,T,4] cxcywh
                        const unsigned char* __restrict__ mask, // [B,T] bool (1B)
                        float* __restrict__ out)             // [3,B,T] flat
{
    __shared__ float4 raw[NP];                      // staged AoS boxes (14.4 KB)
    __shared__ float px0[NP], py0[NP], px1[NP], py1[NP], parea[NP]; // SoA (18 KB)

    const int b   = blockIdx.x;
    const int tid = threadIdx.x;

    const float4* gsrc = (const float4*)(pred + (size_t)b * NP * 4);

    // ---- Stage pred boxes for this batch into LDS (async DMA path) ----
#ifdef HAVE_ASYNC_LDS
    for (int j = tid; j < NP; j += THREADS) {
        __builtin_amdgcn_global_load_async_to_lds_b128(
            (as1_v4i*)(void*)(gsrc + j),
            (as3_v4i*)(void*)(&raw[j]),
            /*imm offset*/0, /*cpol*/0);
    }
#if __has_builtin(__builtin_amdgcn_s_wait_asynccnt)
    __builtin_amdgcn_s_wait_asynccnt(0);
#else
    asm volatile("s_wait_asynccnt 0" ::: "memory");
#endif
#else
    for (int j = tid; j < NP; j += THREADS) raw[j] = gsrc[j];
#endif
    __syncthreads();

    // ---- Convert cxcywh -> xyxy + area, SoA for conflict-free inner loop ----
    for (int j = tid; j < NP; j += THREADS) {
        float4 v = raw[j];
        float hx = 0.5f * v.z, hy = 0.5f * v.w;
        float x0 = v.x - hx, y0 = v.y - hy;
        float x1 = v.x + hx, y1 = v.y + hy;
        px0[j] = x0; py0[j] = y0; px1[j] = x1; py1[j] = y1;
        parea[j] = (x1 - x0) * (y1 - y0);
    }
    __syncthreads();

    // ---- 4 lanes per target, each scans a 225-pred chunk ----
    const int tl  = tid >> 2;        // target within block (0..63)
    const int sub = tid & 3;         // chunk id
    const int t   = blockIdx.y * TGT_PER_BLK + tl;

    float best = -1e30f;
    int   bidx = 0;
    float keep = 0.0f;

    if (t < NT) {
        float4 tb = *(const float4*)(targ + ((size_t)b * NT + t) * 4);
        float thx = 0.5f * tb.z, thy = 0.5f * tb.w;
        float tx0 = tb.x - thx, ty0 = tb.y - thy;
        float tx1 = tb.x + thx, ty1 = tb.y + thy;
        float ta  = (tx1 - tx0) * (ty1 - ty0);

        const int base = sub * PRED_CHUNK;
        #pragma unroll 5
        for (int i = 0; i < PRED_CHUNK; ++i) {
            const int n = base + i;
            const float ax0 = px0[n], ay0 = py0[n];
            const float ax1 = px1[n], ay1 = py1[n];
            // intersection
            float iw = fminf(ax1, tx1) - fmaxf(ax0, tx0);
            float ih = fminf(ay1, ty1) - fmaxf(ay0, ty0);
            iw = fmaxf(iw, 0.0f);
            ih = fmaxf(ih, 0.0f);
            const float inter = iw * ih;
            const float uni   = parea[n] + ta - inter;
            const float iou   = inter * __builtin_amdgcn_rcpf(uni);
            // smallest enclosing box
            float ew = fmaxf(ax1, tx1) - fminf(ax0, tx0);
            float eh = fmaxf(ay1, ty1) - fminf(ay0, ty0);
            ew = fmaxf(ew, 0.0f);
            eh = fmaxf(eh, 0.0f);
            const float ae   = ew * eh;
            const float giou = iou - (ae - uni) * __builtin_amdgcn_rcpf(ae);
            if (giou > best) { best = giou; bidx = n; }  // strict > keeps first max
        }
        keep = 1.0f;
    }

    // ---- combine the 4 chunk-argmaxes (lanes sub^1, sub^2 within wave32) ----
    #pragma unroll
    for (int off = 1; off < 4; off <<= 1) {
        float ov = __shfl_xor(best, off, 32);
        int   oi = __shfl_xor(bidx, off, 32);
        if (ov > best || (ov == best && oi < bidx)) { best = ov; bidx = oi; }
    }

    if (sub == 0 && keep != 0.0f) {
        const int BT = NB * NT;
        const int bt = b * NT + t;
        out[bt]          = (float)bidx;                                    // pred_idx
        out[BT + bt]     = (mask[bt] && best >= 0.5f) ? 1.0f : 0.0f;       // valid
        out[2 * BT + bt] = best;                                           // max_iou
    }
}

extern "C" void kernel_launch(void* const* d_in, const int* in_sizes, int n_in,
                              void* d_out, int out_size, void* d_ws, size_t ws_size,
                              hipStream_t stream) {
    (void)in_sizes; (void)n_in; (void)d_ws; (void)ws_size; (void)out_size;
    const float*         pred = (const float*)d_in[0];
    const float*         targ = (const float*)d_in[1];
    const unsigned char* mask = (const unsigned char*)d_in[2];
    float*               out  = (float*)d_out;

    dim3 grid(NB, (NT + TGT_PER_BLK - 1) / TGT_PER_BLK);
    dim3 block(THREADS);
    giou_argmax_kernel<<<grid, block, 0, stream>>>(pred, targ, mask, out);
}